// Net_77068893159889
// MI455X (gfx1250) — compile-verified
//
#include <hip/hip_runtime.h>
#include <hip/hip_bf16.h>
#include <float.h>
#include <math.h>

// ---------------------------------------------------------------------------
// GraphSAGE (max + mean aggr) fused forward for gfx1250 (MI455X).
//   x:[100000,128] f32, edge_index:[2,1600000] i32, W*:[40,128], b*:[40]
//   out = log_softmax(agg_max@Wl_max^T + x@(Wr_max+Wr_mean)^T
//                     + agg_mean@Wl_mean^T + b_max + b_mean)
// Strategy: L2-resident scatter atomics for aggregation (x/agg arrays = 3*51MB
// < 192MB L2), then fp32 WMMA (V_WMMA_F32_16X16X4_F32) for the [N,384]x[384,48]
// fused GEMM with log-softmax epilogue.
//   - weights stored transposed Bt[48][384] so each lane's (k, k+1) B pair is
//     one global_load_b64 in the WMMA loop
//   - edge indices made wave-uniform via readfirstlane -> scalar loads/SALU
//     address math in the dominant scatter phase
// ---------------------------------------------------------------------------

typedef __attribute__((ext_vector_type(2))) float v2f;
typedef __attribute__((ext_vector_type(8))) float v8f;

#define N_NODES   100000
#define N_EDGES   1600000
#define D_FEAT    128
#define N_CLASSES 40
#define KEXT      384     // 3 * 128 (max | x | mean)
#define NEXT      48      // 40 classes padded to 3 WMMA N-tiles
#define LDS_STR   388     // 388 % 64 == 4 -> bank-conflict-free column reads

// ---------------- init workspace -------------------------------------------
__global__ __launch_bounds__(256) void init_ws_kernel(float* __restrict__ aggmax,
                                                      float* __restrict__ aggsum,
                                                      float* __restrict__ deg) {
  int i = blockIdx.x * blockDim.x + threadIdx.x;
  if (i < N_NODES * D_FEAT) {
    aggmax[i] = -FLT_MAX;
    aggsum[i] = 0.0f;
  }
  if (i < N_NODES) deg[i] = 0.0f;
}

// ---------------- build fused, TRANSPOSED weight matrix Bt[48][384] ---------
__global__ __launch_bounds__(256) void build_bext_kernel(const float* __restrict__ Wl_max,
                                                         const float* __restrict__ Wr_max,
                                                         const float* __restrict__ Wl_mean,
                                                         const float* __restrict__ Wr_mean,
                                                         float* __restrict__ Bt) {
  int i = blockIdx.x * blockDim.x + threadIdx.x;
  if (i >= KEXT * NEXT) return;
  int c = i / KEXT;   // padded class 0..47
  int k = i % KEXT;   // extended feature 0..383
  float v = 0.0f;
  if (c < N_CLASSES) {
    if (k < 128)      v = Wl_max [c * D_FEAT + k];
    else if (k < 256) v = Wr_max [c * D_FEAT + (k - 128)] + Wr_mean[c * D_FEAT + (k - 128)];
    else              v = Wl_mean[c * D_FEAT + (k - 256)];
  }
  Bt[(size_t)c * KEXT + k] = v;
}

// ---------------- edge scatter: one wave32 per edge -------------------------
__global__ __launch_bounds__(256) void edge_scatter_kernel(const float* __restrict__ x,
                                                           const int*   __restrict__ edge_index,
                                                           float* __restrict__ aggmax,
                                                           float* __restrict__ aggsum,
                                                           float* __restrict__ deg) {
  int gid  = blockIdx.x * blockDim.x + threadIdx.x;
  int e    = gid >> 5;          // wave-uniform by construction
  int lane = gid & 31;
  if (e >= N_EDGES) return;
  e = __builtin_amdgcn_readfirstlane(e);   // make uniformity provable -> s_load
  int src = edge_index[e];                 // row 0 (scalar load)
  int dst = edge_index[N_EDGES + e];       // row 1 (scalar load)
  // coalesced 512B gather of x[src] across the wave
  const float4 v = ((const float4*)(x + (size_t)src * D_FEAT))[lane];
  float* pm = aggmax + (size_t)dst * D_FEAT + lane * 4;
  float* ps = aggsum + (size_t)dst * D_FEAT + lane * 4;
  atomicMax(pm + 0, v.x); atomicMax(pm + 1, v.y);
  atomicMax(pm + 2, v.z); atomicMax(pm + 3, v.w);
  atomicAdd(ps + 0, v.x); atomicAdd(ps + 1, v.y);
  atomicAdd(ps + 2, v.z); atomicAdd(ps + 3, v.w);
  if (lane == 0) atomicAdd(deg + dst, 1.0f);
}

// ---------------- WMMA GEMM + log-softmax: one wave per 16-node tile --------
__global__ __launch_bounds__(32) void sage_wmma_kernel(const float* __restrict__ x,
                                                       const float* __restrict__ aggmax,
                                                       const float* __restrict__ aggsum,
                                                       const float* __restrict__ deg,
                                                       const float* __restrict__ Bt,
                                                       const float* __restrict__ b_max,
                                                       const float* __restrict__ b_mean,
                                                       float* __restrict__ out) {
  __shared__ float As[16 * LDS_STR];   // 24832 B: A-tile [16 nodes][384 feats], later reused for D
  const int lane  = threadIdx.x;
  const int node0 = blockIdx.x * 16;

  // ---- stage A-tile into LDS (each lane: one float4 per 128-float segment) ----
  for (int m = 0; m < 16; ++m) {
    int node = node0 + m;
    float4 vmax, vx, vmean;
    if (node < N_NODES) {
      float d    = deg[node];
      float invd = 1.0f / fmaxf(d, 1.0f);
      vmax  = ((const float4*)(aggmax + (size_t)node * D_FEAT))[lane];
      vx    = ((const float4*)(x      + (size_t)node * D_FEAT))[lane];
      vmean = ((const float4*)(aggsum + (size_t)node * D_FEAT))[lane];
      vmean.x *= invd; vmean.y *= invd; vmean.z *= invd; vmean.w *= invd;
      if (d == 0.0f) { vmax = make_float4(0.f, 0.f, 0.f, 0.f); }  // isolated: -inf -> 0
    } else {
      vmax = vx = vmean = make_float4(0.f, 0.f, 0.f, 0.f);
    }
    float* row = As + m * LDS_STR;
    row[lane * 4 + 0]       = vmax.x;  row[lane * 4 + 1]       = vmax.y;
    row[lane * 4 + 2]       = vmax.z;  row[lane * 4 + 3]       = vmax.w;
    row[128 + lane * 4 + 0] = vx.x;    row[128 + lane * 4 + 1] = vx.y;
    row[128 + lane * 4 + 2] = vx.z;    row[128 + lane * 4 + 3] = vx.w;
    row[256 + lane * 4 + 0] = vmean.x; row[256 + lane * 4 + 1] = vmean.y;
    row[256 + lane * 4 + 2] = vmean.z; row[256 + lane * 4 + 3] = vmean.w;
  }
  __syncthreads();

  // ---- K-loop: V_WMMA_F32_16X16X4_F32, 96 steps x 3 N-tiles ----
  // A 16x4 fp32 layout: lanes 0-15 hold M=lane, VGPR0/1 = K0,K1;
  //                     lanes 16-31 hold M=lane-16, VGPR0/1 = K2,K3.
  // B 4x16 fp32 layout: lanes 0-15 hold N=lane, VGPR0/1 = K0,K1;
  //                     lanes 16-31 hold N=lane-16, VGPR0/1 = K2,K3.
  const int arow = lane & 15;
  const int koff = (lane >> 4) * 2;
  const float* Bp0 = Bt + (size_t)(arow +  0) * KEXT + koff;  // contiguous (k,k+1) pairs
  const float* Bp1 = Bt + (size_t)(arow + 16) * KEXT + koff;
  const float* Bp2 = Bt + (size_t)(arow + 32) * KEXT + koff;
  v8f c0 = {}; v8f c1 = {}; v8f c2 = {};
  for (int kt = 0; kt < KEXT / 4; ++kt) {
    const int k0 = kt * 4;
    v2f a;
    a.x = As[arow * LDS_STR + k0 + koff];
    a.y = As[arow * LDS_STR + k0 + koff + 1];
    const v2f b0 = *(const v2f*)(Bp0 + k0);   // global_load_b64
    const v2f b1 = *(const v2f*)(Bp1 + k0);
    const v2f b2 = *(const v2f*)(Bp2 + k0);
    c0 = __builtin_amdgcn_wmma_f32_16x16x4_f32(false, a, false, b0, (short)0, c0, false, false);
    c1 = __builtin_amdgcn_wmma_f32_16x16x4_f32(false, a, false, b1, (short)0, c1, false, false);
    c2 = __builtin_amdgcn_wmma_f32_16x16x4_f32(false, a, false, b2, (short)0, c2, false, false);
  }
  __syncthreads();

  // ---- spill D to LDS as Ld[16 nodes][48 classes] ----
  // D layout: VGPR r, lanes 0-15 -> (M=r, N=lane), lanes 16-31 -> (M=r+8, N=lane-16)
  float* Ld = As;
  {
    const int mbase = (lane >> 4) * 8;
    const int ncol  = lane & 15;
    for (int r = 0; r < 8; ++r) {
      Ld[(mbase + r) * NEXT + ncol +  0] = c0[r];
      Ld[(mbase + r) * NEXT + ncol + 16] = c1[r];
      Ld[(mbase + r) * NEXT + ncol + 32] = c2[r];
    }
  }
  __syncthreads();

  // ---- per-node bias + log-softmax (lanes 0-15 each own a node row) ----
  if (lane < 16) {
    int node = node0 + lane;
    if (node < N_NODES) {
      float* row = Ld + lane * NEXT;
      float mx = -FLT_MAX;
      for (int c = 0; c < N_CLASSES; ++c) {
        float v = row[c] + b_max[c] + b_mean[c];
        row[c] = v;
        mx = fmaxf(mx, v);
      }
      float s = 0.0f;
      for (int c = 0; c < N_CLASSES; ++c) s += expf(row[c] - mx);
      float lse = mx + logf(s);
      float* o = out + (size_t)node * N_CLASSES;
      for (int c = 0; c < N_CLASSES; ++c) o[c] = row[c] - lse;
    }
  }
}

// ---------------------------------------------------------------------------
extern "C" void kernel_launch(void* const* d_in, const int* in_sizes, int n_in,
                              void* d_out, int out_size, void* d_ws, size_t ws_size,
                              hipStream_t stream) {
  const float* x        = (const float*)d_in[0];
  const int*   ei       = (const int*)  d_in[1];
  const float* Wl_max   = (const float*)d_in[2];
  const float* Wr_max   = (const float*)d_in[3];
  const float* b_max    = (const float*)d_in[4];
  const float* Wl_mean  = (const float*)d_in[5];
  const float* Wr_mean  = (const float*)d_in[6];
  const float* b_mean   = (const float*)d_in[7];
  float* out = (float*)d_out;

  // workspace carve-up (~103 MB)
  float* aggmax = (float*)d_ws;                       // 12,800,000 f32
  float* aggsum = aggmax + (size_t)N_NODES * D_FEAT;  // 12,800,000 f32
  float* degw   = aggsum + (size_t)N_NODES * D_FEAT;  //    100,000 f32
  float* Bt     = degw + N_NODES;                     //     18,432 f32

  {
    int total = N_NODES * D_FEAT;
    init_ws_kernel<<<(total + 255) / 256, 256, 0, stream>>>(aggmax, aggsum, degw);
  }
  {
    int total = KEXT * NEXT;
    build_bext_kernel<<<(total + 255) / 256, 256, 0, stream>>>(Wl_max, Wr_max, Wl_mean, Wr_mean, Bt);
  }
  {
    long long total = (long long)N_EDGES * 32;
    edge_scatter_kernel<<<(int)((total + 255) / 256), 256, 0, stream>>>(x, ei, aggmax, aggsum, degw);
  }
  {
    int tiles = (N_NODES + 15) / 16;  // 6250
    sage_wmma_kernel<<<tiles, 32, 0, stream>>>(x, aggmax, aggsum, degw, Bt, b_max, b_mean, out);
  }
}